// SSM_43636867727959
// MI455X (gfx1250) — compile-verified
//
#include <hip/hip_runtime.h>

#define BATCH 64
#define SEQ   512
#define DI    64
#define DSN   64
#define DTRK  4
#define XPJ   132   // DT_RANK + 2*D_STATE

typedef float v2f __attribute__((ext_vector_type(2)));
typedef float v8f __attribute__((ext_vector_type(8)));

// D = A(16x4) * B(4x16) + C, fp32 WMMA (wave32)
__device__ __forceinline__ v8f wmma4(v2f a, v2f b, v8f c) {
  return __builtin_amdgcn_wmma_f32_16x16x4_f32(false, a, false, b, (short)0, c,
                                               false, false);
}

// C(64x64, LDS) = A(64x64, LDS) @ B(64x64, LDS); 16 tiles over 8 waves
__device__ __forceinline__ void mm64_lll(const float* A, const float* B,
                                         float* C, int tid) {
  const int wave = tid >> 5, lane = tid & 31;
  const int lm = lane & 15;
  const int koff = (lane >> 4) << 1;  // lanes 16-31 hold K+2,K+3
  const int mo = (lane >> 4) << 3;    // D rows: lanes 16-31 are M=r+8
  for (int tile = wave; tile < 16; tile += 8) {
    const int tm = (tile >> 2) << 4, tn = (tile & 3) << 4;
    v8f acc = {};
#pragma unroll
    for (int kb = 0; kb < 64; kb += 4) {
      v2f a, b;
      a.x = A[(tm + lm) * 64 + kb + koff];
      a.y = A[(tm + lm) * 64 + kb + koff + 1];
      b.x = B[(kb + koff) * 64 + tn + lm];
      b.y = B[(kb + koff + 1) * 64 + tn + lm];
      acc = wmma4(a, b, acc);
    }
#pragma unroll
    for (int r = 0; r < 8; ++r) C[(tm + r + mo) * 64 + tn + lm] = acc[r];
  }
}

// Cg(64x64, global) = A(64x64, LDS) @ Bg(64x64, global)
__device__ __forceinline__ void mm64_lgg(const float* A,
                                         const float* __restrict__ Bg,
                                         float* __restrict__ Cg, int tid) {
  const int wave = tid >> 5, lane = tid & 31;
  const int lm = lane & 15;
  const int koff = (lane >> 4) << 1;
  const int mo = (lane >> 4) << 3;
  for (int tile = wave; tile < 16; tile += 8) {
    const int tm = (tile >> 2) << 4, tn = (tile & 3) << 4;
    v8f acc = {};
#pragma unroll
    for (int kb = 0; kb < 64; kb += 4) {
      v2f a, b;
      a.x = A[(tm + lm) * 64 + kb + koff];
      a.y = A[(tm + lm) * 64 + kb + koff + 1];
      b.x = Bg[(kb + koff) * 64 + tn + lm];
      b.y = Bg[(kb + koff + 1) * 64 + tn + lm];
      acc = wmma4(a, b, acc);
    }
#pragma unroll
    for (int r = 0; r < 8; ++r) Cg[(tm + r + mo) * 64 + tn + lm] = acc[r];
  }
}

// One block per timestep t: compute A_bar[t], B_bar[t] (and C at t=SEQ-1).
__global__ __launch_bounds__(256) void ssm_coeff_kernel(
    const float* __restrict__ x, const float* __restrict__ Wxp,
    const float* __restrict__ bxp, const float* __restrict__ Wdt,
    const float* __restrict__ bdt, const float* __restrict__ Alog,
    float* __restrict__ Abar, float* __restrict__ Bbar,
    float* __restrict__ Gws, float* __restrict__ Cws) {
  __shared__ float S[16384];           // exactly 64 KB, regions overlaid
  float* Xs = S;                       // [0,4096): x_t (64x64)
  float* XPs = S + 4096;               // [4096,12544): xp (64x132)
  float* M = S;                        // matexp input (over Xs)
  float* E = S + 4096;                 // e^M accumulator (over XPs)
  float* P = S + 8192;                 // phi1 accumulator
  float* T = S + 12288;                // matmul temp
  float* RED = S + 4096;               // transient reduce (before E init)

  const int t = blockIdx.x;
  const int tid = threadIdx.x;
  const int wave = tid >> 5, lane = tid & 31;
  const int lm = lane & 15;
  const int koff = (lane >> 4) << 1;
  const int mo = (lane >> 4) << 3;

  // ---- stage x_t into LDS ----
  for (int i = tid; i < 4096; i += 256) {
    const int b = i >> 6, d = i & 63;
    Xs[i] = x[((size_t)b * SEQ + t) * DI + d];
  }
  __syncthreads();

  // ---- xp = x_t @ Wxp^T + bxp : M=64, N=132 (pad 144), K=64 ----
  for (int tile = wave; tile < 36; tile += 8) {
    const int tm = (tile / 9) << 4;
    const int tn = (tile % 9) << 4;
    const int j = tn + lm;
    const float jmask = (j < XPJ) ? 1.f : 0.f;
    const int jc = (j < XPJ) ? j : 0;
    v8f acc = {};
#pragma unroll
    for (int kb = 0; kb < 64; kb += 4) {
      v2f a, b;
      a.x = Xs[(tm + lm) * 64 + kb + koff];
      a.y = Xs[(tm + lm) * 64 + kb + koff + 1];
      b.x = Wxp[jc * DI + kb + koff] * jmask;
      b.y = Wxp[jc * DI + kb + koff + 1] * jmask;
      acc = wmma4(a, b, acc);
    }
    const float bias = bxp[jc] * jmask;
    if (j < XPJ) {
#pragma unroll
      for (int r = 0; r < 8; ++r)
        XPs[(tm + r + mo) * XPJ + j] = acc[r] + bias;
    }
  }
  __syncthreads();

  // ---- final step: save C = xp[:, 68:132] for the y computation ----
  if (t == SEQ - 1) {
    for (int i = tid; i < 4096; i += 256) {
      const int b = i >> 6, n = i & 63;
      Cws[i] = XPs[b * XPJ + 68 + n];
    }
  }

  // ---- dt = dtr @ Wdt^T + bdt;  At -> M (LDS);  G = dt*Bm -> global ----
  for (int tile = wave; tile < 16; tile += 8) {
    const int tm = (tile >> 2) << 4, tn = (tile & 3) << 4;
    v2f a, b;
    a.x = XPs[(tm + lm) * XPJ + 64 + koff];
    a.y = XPs[(tm + lm) * XPJ + 64 + koff + 1];
    b.x = Wdt[(tn + lm) * DTRK + koff];       // Wdt^T[k][n] = Wdt[n*4+k]
    b.y = Wdt[(tn + lm) * DTRK + koff + 1];
    v8f acc = {};
    acc = wmma4(a, b, acc);
#pragma unroll
    for (int r = 0; r < 8; ++r) {
      const int i = tm + r + mo;              // row (batch index)
      const int j = tn + lm;                  // col (state index)
      const float dtv = acc[r] + bdt[j];
      const float Aij = -__expf(Alog[i * DSN + j]);
      M[i * 64 + j] = dtv * Aij;                                  // At
      Gws[(size_t)t * 4096 + i * 64 + j] = dtv * XPs[i * XPJ + j]; // dt*Bm
    }
  }
  __syncthreads();

  // ---- inf-norm of At -> scaling count sc ----
  if (tid < 64) {
    float s = 0.f;
    for (int j = 0; j < 64; ++j) s += fabsf(M[tid * 64 + j]);
    RED[tid] = s;
  }
  __syncthreads();
  if (tid == 0) {
    float mx = 0.f;
    for (int r = 0; r < 64; ++r) mx = fmaxf(mx, RED[r]);
    int scv;
    if (!(mx < 1e30f)) scv = 48;                      // inf/NaN guard
    else if (mx <= 0.25f) scv = 0;
    else scv = (int)ceilf(log2f(mx * 4.0f));          // ceil(log2(mx/0.25))
    if (scv < 0) scv = 0;
    if (scv > 48) scv = 48;
    RED[64] = (float)scv;
  }
  __syncthreads();
  const int sc = (int)RED[64];
  __syncthreads();  // everyone captured sc before E overwrites RED region
  const float scale = ldexpf(1.f, -sc);
  for (int i = tid; i < 4096; i += 256) M[i] *= scale;
  __syncthreads();

  // ---- Taylor-Horner order 8: E = exp(M) ----
  for (int i = tid; i < 4096; i += 256) E[i] = (i % 65 == 0) ? 1.f : 0.f;
  __syncthreads();
#pragma unroll 1
  for (int k = 8; k >= 1; --k) {
    mm64_lll(M, E, T, tid);
    __syncthreads();
    const float inv = 1.f / (float)k;
    for (int i = tid; i < 4096; i += 256)
      E[i] = T[i] * inv + ((i % 65 == 0) ? 1.f : 0.f);
    __syncthreads();
  }

  // ---- Taylor-Horner order 8: P = phi1(M) = sum M^k/(k+1)! ----
  float c = 1.f / 362880.f;  // 1/9!
  for (int i = tid; i < 4096; i += 256) P[i] = (i % 65 == 0) ? c : 0.f;
  __syncthreads();
#pragma unroll 1
  for (int k = 7; k >= 0; --k) {
    c = c * (float)(k + 2);  // c = 1/(k+1)!
    mm64_lll(M, P, T, tid);
    __syncthreads();
    for (int i = tid; i < 4096; i += 256)
      P[i] = T[i] + ((i % 65 == 0) ? c : 0.f);
    __syncthreads();
  }

  // ---- squaring: E<-E^2, P<-(E@P+P)/2, sc times (M is free temp) ----
#pragma unroll 1
  for (int sq = 0; sq < sc; ++sq) {
    mm64_lll(E, P, T, tid);
    __syncthreads();
    mm64_lll(E, E, M, tid);
    __syncthreads();
    for (int i = tid; i < 4096; i += 256) {
      P[i] = 0.5f * (T[i] + P[i]);
      E[i] = M[i];
    }
    __syncthreads();
  }

  // ---- store A_bar; B_bar = phi1(At) @ (dt*Bm) straight to global ----
  for (int i = tid; i < 4096; i += 256) Abar[(size_t)t * 4096 + i] = E[i];
  mm64_lgg(P, Gws + (size_t)t * 4096, Bbar + (size_t)t * 4096, tid);
}

// Elementwise recurrence; one block per (batch, n-quarter). Streams hs (512MB).
__global__ __launch_bounds__(256) void ssm_scan_kernel(
    const float* __restrict__ x, const float* __restrict__ Abar,
    const float* __restrict__ Bbar, const float* __restrict__ Cws,
    float* __restrict__ hs, float* __restrict__ ypart) {
  const int b = blockIdx.x >> 2;
  const int q = blockIdx.x & 3;
  const int tid = threadIdx.x;
  const int d = tid >> 2;
  const int n0 = (q << 4) + ((tid & 3) << 2);
  const int cofs = d * 64 + n0;
  const float* xb = x + (size_t)b * SEQ * DI + d;
  float* hout = hs + (size_t)b * SEQ * 4096 + cofs;
  float4 h = make_float4(0.f, 0.f, 0.f, 0.f);
#pragma unroll 1
  for (int t = 0; t < SEQ; ++t) {
    const float xv = xb[(size_t)t * DI];
    const float4 a = *(const float4*)(Abar + (size_t)t * 4096 + cofs);
    const float4 bb = *(const float4*)(Bbar + (size_t)t * 4096 + cofs);
    h.x = fminf(fmaxf(h.x * a.x + xv * bb.x, -1e6f), 1e6f);
    h.y = fminf(fmaxf(h.y * a.y + xv * bb.y, -1e6f), 1e6f);
    h.z = fminf(fmaxf(h.z * a.z + xv * bb.z, -1e6f), 1e6f);
    h.w = fminf(fmaxf(h.w * a.w + xv * bb.w, -1e6f), 1e6f);
    *(float4*)(hout + (size_t)t * 4096) = h;
    if (t + 1 < SEQ) {  // pull next step's coefficients (global_prefetch_b8)
      __builtin_prefetch(Abar + (size_t)(t + 1) * 4096 + cofs, 0, 1);
      __builtin_prefetch(Bbar + (size_t)(t + 1) * 4096 + cofs, 0, 1);
    }
  }
  // partial y[b,d] over this block's 16 n-values
  const float4 cc = *(const float4*)(Cws + b * 64 + n0);
  const float part = h.x * cc.x + h.y * cc.y + h.z * cc.z + h.w * cc.w;
  __shared__ float red[256];
  red[tid] = part;
  __syncthreads();
  if ((tid & 3) == 0)
    ypart[blockIdx.x * 64 + d] =
        red[tid] + red[tid + 1] + red[tid + 2] + red[tid + 3];
}

__global__ __launch_bounds__(256) void ssm_y_kernel(
    const float* __restrict__ x, const float* __restrict__ Dv,
    const float* __restrict__ ypart, float* __restrict__ y) {
  const int idx = blockIdx.x * 256 + threadIdx.x;  // 4096 = BATCH*DI
  const int b = idx >> 6, d = idx & 63;
  float s = 0.f;
#pragma unroll
  for (int q = 0; q < 4; ++q) s += ypart[((b << 2) + q) * 64 + d];
  y[idx] = s + Dv[d] * x[((size_t)b * SEQ + (SEQ - 1)) * DI + d];
}

extern "C" void kernel_launch(void* const* d_in, const int* in_sizes, int n_in,
                              void* d_out, int out_size, void* d_ws,
                              size_t ws_size, hipStream_t stream) {
  const float* x = (const float*)d_in[0];
  const float* Wxp = (const float*)d_in[1];
  const float* bxp = (const float*)d_in[2];
  const float* Wdt = (const float*)d_in[3];
  const float* bdt = (const float*)d_in[4];
  const float* Alog = (const float*)d_in[5];
  const float* Dv = (const float*)d_in[6];

  float* ws = (float*)d_ws;
  float* Abar = ws;                              // 512*4096
  float* Bbar = Abar + (size_t)SEQ * 4096;       // 512*4096
  float* Gws = Bbar + (size_t)SEQ * 4096;        // 512*4096
  float* Cws = Gws + (size_t)SEQ * 4096;         // 4096
  float* ypart = Cws + 4096;                     // 16384

  float* y = (float*)d_out;                      // ys[-1]: 4096 floats
  float* hs = y + 4096;                          // hs: 64*512*64*64 floats

  hipLaunchKernelGGL(ssm_coeff_kernel, dim3(SEQ), dim3(256), 0, stream, x, Wxp,
                     bxp, Wdt, bdt, Alog, Abar, Bbar, Gws, Cws);
  hipLaunchKernelGGL(ssm_scan_kernel, dim3(BATCH * 4), dim3(256), 0, stream, x,
                     Abar, Bbar, Cws, hs, ypart);
  hipLaunchKernelGGL(ssm_y_kernel, dim3(16), dim3(256), 0, stream, x, Dv,
                     ypart, y);
}